// SGNet01_60687887893296
// MI455X (gfx1250) — compile-verified
//
#include <hip/hip_runtime.h>
#include <hip/hip_bf16.h>
#include <stdint.h>

#define BLOCK 256
#define TILE  1024          // edges staged per LDS tile (4 KB per int/float array)
#define EPT   (TILE/BLOCK)  // 4 edges per thread

typedef int   v4i __attribute__((ext_vector_type(4)));
typedef float v4f __attribute__((ext_vector_type(4)));

// ---- CDNA5 async global->LDS helpers (ASYNCcnt path) -----------------------
__device__ __forceinline__ uint32_t lds_off(const void* p) {
    // generic pointer to LDS: low 32 bits are the LDS byte address (ISA 10.2)
    return (uint32_t)(uintptr_t)p;
}
__device__ __forceinline__ void async_copy_b128(uint32_t lds_addr, const void* gptr) {
    unsigned long long ga = (unsigned long long)(uintptr_t)gptr;
    asm volatile("global_load_async_to_lds_b128 %0, %1, off"
                 :: "v"(lds_addr), "v"(ga) : "memory");
}
__device__ __forceinline__ void async_wait_le(int n) {
    if (n == 0) asm volatile("s_wait_asynccnt 0" ::: "memory");
    else        asm volatile("s_wait_asynccnt 3" ::: "memory");
}

// ---- node-level kernels ----------------------------------------------------
__global__ void k_init_deg(float* __restrict__ deg, int N) {
    int i = blockIdx.x * blockDim.x + threadIdx.x;
    if (i < N) deg[i] = 1.0f;                    // self-loop weight
}

// 128-bit nontemporal streaming of the one-touch degree pass
__global__ void k_accum_deg(const int* __restrict__ col, const float* __restrict__ w,
                            float* __restrict__ deg, long long E, long long nV) {
    long long stride = (long long)gridDim.x * blockDim.x;
    const v4i* c4 = (const v4i*)col;
    const v4f* w4 = (const v4f*)w;
    for (long long v = (long long)blockIdx.x * blockDim.x + threadIdx.x; v < nV; v += stride) {
        v4i c  = __builtin_nontemporal_load(&c4[v]);
        v4f wv = __builtin_nontemporal_load(&w4[v]);
        atomicAdd(&deg[c.x], wv.x);
        atomicAdd(&deg[c.y], wv.y);
        atomicAdd(&deg[c.z], wv.z);
        atomicAdd(&deg[c.w], wv.w);
    }
    for (long long e = (nV << 2) + (long long)blockIdx.x * blockDim.x + threadIdx.x;
         e < E; e += stride) {
        atomicAdd(&deg[col[e]], w[e]);           // tail / unaligned fallback
    }
}

__global__ void k_dinv(const float* __restrict__ deg, float* __restrict__ dinv, int N) {
    int i = blockIdx.x * blockDim.x + threadIdx.x;
    if (i < N) {
        float d = deg[i];
        dinv[i] = d > 0.0f ? rsqrtf(d) : 0.0f;
    }
}

// pack node state {dinv, h} for single-b64 gathers AND write self-loop term
__global__ void k_pack(const float* __restrict__ src, const float* __restrict__ dinv,
                       float2* __restrict__ pk, float* __restrict__ dst, int N) {
    int i = blockIdx.x * blockDim.x + threadIdx.x;
    if (i < N) {
        float di = dinv[i], s = src[i];
        pk[i]  = make_float2(di, s);
        dst[i] = di * di * s;
    }
}

// fallback self-loop init (unpacked path)
__global__ void k_self(const float* __restrict__ src, const float* __restrict__ dinv,
                       float* __restrict__ dst, int N) {
    int i = blockIdx.x * blockDim.x + threadIdx.x;
    if (i < N) {
        float di = dinv[i];
        dst[i] = di * di * src[i];
    }
}

// ---- edge propagation: async-to-LDS double-buffered edge stream ------------
// packed variant: 2 gathers/edge (pk[r] as b64, pk[c].x as b32)
__global__ void k_prop_pk(const int* __restrict__ row, const int* __restrict__ col,
                          const float* __restrict__ w, const float2* __restrict__ pk,
                          float* __restrict__ dst, long long E) {
    __shared__ __align__(16) int   s_row[2][TILE];
    __shared__ __align__(16) int   s_col[2][TILE];
    __shared__ __align__(16) float s_w  [2][TILE];

    const int t = threadIdx.x;
    const long long nFull = E / TILE;

    auto stage = [&](int b, long long tile) {
        long long base = tile * (long long)TILE;
        int e0 = t * 4;                                   // 16 B per lane per array
        async_copy_b128(lds_off(&s_row[b][e0]), row + base + e0);
        async_copy_b128(lds_off(&s_col[b][e0]), col + base + e0);
        async_copy_b128(lds_off(&s_w  [b][e0]), w   + base + e0);
    };

    long long tile = blockIdx.x;
    int pb = 0;
    if (tile < nFull) stage(pb, tile);
    while (tile < nFull) {
        long long nxt = tile + gridDim.x;
        if (nxt < nFull) { stage(pb ^ 1, nxt); async_wait_le(3); }  // next tile in flight
        else             { async_wait_le(0); }
        __syncthreads();                                  // all waves' tiles landed
        #pragma unroll
        for (int k = 0; k < EPT; ++k) {
            int e = t + k * BLOCK;                        // conflict-free LDS reads
            int   r  = s_row[pb][e];
            int   c  = s_col[pb][e];
            float wv = s_w  [pb][e];
            float2 pr = pk[r];                            // {dinv[r], h[r]} one b64 gather
            float  dc = pk[c].x;                          // dinv[c]
            atomicAdd(&dst[c], pr.x * wv * dc * pr.y);    // L2-resident scatter
        }
        __syncthreads();                                  // done reading before re-stage
        pb ^= 1;
        tile = nxt;
    }

    long long tail0 = nFull * (long long)TILE;
    long long gstride = (long long)gridDim.x * blockDim.x;
    for (long long e = tail0 + (long long)blockIdx.x * blockDim.x + t; e < E; e += gstride) {
        int r = row[e], c = col[e];
        float2 pr = pk[r];
        atomicAdd(&dst[c], pr.x * w[e] * pk[c].x * pr.y);
    }
}

// fallback variant: separate dinv/src tables (3 gathers/edge)
__global__ void k_prop(const int* __restrict__ row, const int* __restrict__ col,
                       const float* __restrict__ w, const float* __restrict__ dinv,
                       const float* __restrict__ src, float* __restrict__ dst,
                       long long E) {
    __shared__ __align__(16) int   s_row[2][TILE];
    __shared__ __align__(16) int   s_col[2][TILE];
    __shared__ __align__(16) float s_w  [2][TILE];

    const int t = threadIdx.x;
    const long long nFull = E / TILE;

    auto stage = [&](int b, long long tile) {
        long long base = tile * (long long)TILE;
        int e0 = t * 4;
        async_copy_b128(lds_off(&s_row[b][e0]), row + base + e0);
        async_copy_b128(lds_off(&s_col[b][e0]), col + base + e0);
        async_copy_b128(lds_off(&s_w  [b][e0]), w   + base + e0);
    };

    long long tile = blockIdx.x;
    int pb = 0;
    if (tile < nFull) stage(pb, tile);
    while (tile < nFull) {
        long long nxt = tile + gridDim.x;
        if (nxt < nFull) { stage(pb ^ 1, nxt); async_wait_le(3); }
        else             { async_wait_le(0); }
        __syncthreads();
        #pragma unroll
        for (int k = 0; k < EPT; ++k) {
            int e = t + k * BLOCK;
            int   r  = s_row[pb][e];
            int   c  = s_col[pb][e];
            float wv = s_w  [pb][e];
            atomicAdd(&dst[c], dinv[r] * wv * dinv[c] * src[r]);
        }
        __syncthreads();
        pb ^= 1;
        tile = nxt;
    }

    long long tail0 = nFull * (long long)TILE;
    long long gstride = (long long)gridDim.x * blockDim.x;
    for (long long e = tail0 + (long long)blockIdx.x * blockDim.x + t; e < E; e += gstride) {
        int r = row[e], c = col[e];
        atomicAdd(&dst[c], dinv[r] * w[e] * dinv[c] * src[r]);
    }
}

__global__ void k_final(const float* __restrict__ h,
                        const float* __restrict__ cw, const float* __restrict__ cb,
                        const float* __restrict__ lw, const float* __restrict__ lb,
                        float* __restrict__ out, int N) {
    int i = blockIdx.x * blockDim.x + threadIdx.x;
    if (i < N) {
        float z = fmaf(h[i], cw[0], cb[0]);   // SGConv 1x1 linear
        z = fmaxf(z, 0.0f);                   // relu
        z = fmaf(z, lw[0], lb[0]);            // endLinear 1x1
        out[i] = 1.0f / (1.0f + expf(-z));    // sigmoid
    }
}

// ---- launch ----------------------------------------------------------------
extern "C" void kernel_launch(void* const* d_in, const int* in_sizes, int n_in,
                              void* d_out, int out_size, void* d_ws, size_t ws_size,
                              hipStream_t stream) {
    const float* x  = (const float*)d_in[0];
    const int*   ei = (const int*)  d_in[1];   // (2, E): row = ei[0:E), col = ei[E:2E)
    const float* ew = (const float*)d_in[2];
    const float* cw = (const float*)d_in[3];
    const float* cb = (const float*)d_in[4];
    const float* lw = (const float*)d_in[5];
    const float* lb = (const float*)d_in[6];

    const int       N = in_sizes[0];           // x is (N,1)
    const long long E = (long long)in_sizes[2];
    const int* row = ei;
    const int* col = ei + E;

    float* ws   = (float*)d_ws;
    float* deg  = ws;                          // reused as h2 after dinv is computed
    float* dinv = ws + N;
    float* h1   = ws + 2LL * N;
    float* h2   = deg;
    float* out  = (float*)d_out;

    const int nb = (N + BLOCK - 1) / BLOCK;
    const int eb = 2048;                       // grid-stride blocks for edge passes

    // 128-bit vector path only if col/w stay 16B-aligned (E % 4 == 0)
    const long long nV = ((E & 3) == 0) ? (E >> 2) : 0;

    k_init_deg <<<nb, BLOCK, 0, stream>>>(deg, N);
    k_accum_deg<<<eb, BLOCK, 0, stream>>>(col, ew, deg, E, nV);
    k_dinv     <<<nb, BLOCK, 0, stream>>>(deg, dinv, N);

    const size_t need_packed = (size_t)5 * (size_t)N * sizeof(float);
    if (ws_size >= need_packed) {
        float2* pk = (float2*)(ws + 3LL * N);  // {dinv, h} packed node table (2N floats)

        k_pack   <<<nb, BLOCK, 0, stream>>>(x, dinv, pk, h1, N);    // hop 1
        k_prop_pk<<<eb, BLOCK, 0, stream>>>(row, col, ew, pk, h1, E);

        k_pack   <<<nb, BLOCK, 0, stream>>>(h1, dinv, pk, h2, N);   // hop 2
        k_prop_pk<<<eb, BLOCK, 0, stream>>>(row, col, ew, pk, h2, E);
    } else {
        k_self   <<<nb, BLOCK, 0, stream>>>(x, dinv, h1, N);        // hop 1
        k_prop   <<<eb, BLOCK, 0, stream>>>(row, col, ew, dinv, x, h1, E);

        k_self   <<<nb, BLOCK, 0, stream>>>(h1, dinv, h2, N);       // hop 2
        k_prop   <<<eb, BLOCK, 0, stream>>>(row, col, ew, dinv, h1, h2, E);
    }

    k_final    <<<nb, BLOCK, 0, stream>>>(h2, cw, cb, lw, lb, out, N);
}